// Cooc_Layer_5171140624449
// MI455X (gfx1250) — compile-verified
//
#include <hip/hip_runtime.h>
#include <hip/hip_bf16.h>

typedef __attribute__((ext_vector_type(2))) float v2f;
typedef __attribute__((ext_vector_type(8))) float v8f;

#define BATCH 64
#define CH    512
#define HW    784          // 28*28
#define PIXGROUPS 3136     // 64 * (784/16)
#define WAVES_PER_BLOCK 8

// One wave32 handles one group of 16 pixels for all 512 channels.
// g[c] = sum_{k<c} s[k] + sum_{k>c} t[k],  s[k]=x[k]*w[k], t[k]=x[k]*w[k-1]
// In-tile (16-channel) exclusive prefix/suffix scans via V_WMMA_F32_16X16X4_F32
// with constant strictly-triangular A operands synthesized from the lane id.
// Cross-tile carries: per-tile sums staged in LDS + two scalar running sums.
__global__ __launch_bounds__(256) void cooc_wmma_kernel(
    const float* __restrict__ x, const float* __restrict__ w,
    float* __restrict__ g, float* __restrict__ partials)
{
  // wpad[k] = w[k-1], zero-padded both ends: s[k]=x[k]*wpad[k+1], t[k]=x[k]*wpad[k]
  __shared__ float wpad[516];
  __shared__ float Stile[WAVES_PER_BLOCK][32][16];
  __shared__ float Ttile[WAVES_PER_BLOCK][32][16];
  for (int k = (int)threadIdx.x; k < 516; k += 256)
    wpad[k] = (k >= 1 && k <= 511) ? w[k - 1] : 0.0f;
  __syncthreads();

  const int lane = threadIdx.x & 31;
  const int hi   = (lane >> 4) & 1;      // 0: lanes 0-15, 1: lanes 16-31
  const int n    = lane & 15;            // pixel (column) index within group
  const int wave = threadIdx.x >> 5;
  const int G    = blockIdx.x * WAVES_PER_BLOCK + wave;   // 0..3135
  const int b    = G / 49;
  const int pg   = G % 49;
  const int p    = pg * 16 + n;          // pixel index within batch image

  const float* xb = x + (size_t)b * CH * HW + p;
  float*       gb = g + (size_t)b * CH * HW + p;

  // Per-lane base pointers folding in the lane-half K/row offsets, so all
  // inner-loop offsets are compile-time immediates.
  const float* xA = xb + (hi ? 2 : 0) * HW;   // B-fragment loads (K rows)
  const float* xL = xb + (hi ? 8 : 0) * HW;   // loss reload (D rows i / i+8)
  float*       gL = gb + (hi ? 8 : 0) * HW;   // D stores

  // Constant triangular A fragments, one per K-step kk (K = 4*kk + {0..3}).
  // A 16x4 layout: row m = lane&15; VGPR0 holds K (0 lanes<16 / 2 lanes>=16), VGPR1 = K+1.
  const int m = n;
  v2f aL[4], aU[4];
  #pragma unroll
  for (int kk = 0; kk < 4; ++kk) {
    int k0 = kk * 4 + (hi ? 2 : 0);
    aL[kk].x = (k0     < m) ? 1.0f : 0.0f;   // strictly lower: prefix of s
    aL[kk].y = (k0 + 1 < m) ? 1.0f : 0.0f;
    aU[kk].x = (k0     > m) ? 1.0f : 0.0f;   // strictly upper: suffix of t
    aU[kk].y = (k0 + 1 > m) ? 1.0f : 0.0f;
  }

  // ---- Pass A (HBM stream): per-16-channel-tile sums of s and t -> LDS ----
  float totalT = 0.f;
  #pragma unroll 1
  for (int j = 0; j < 32; ++j) {
    const float* xj = xA + (size_t)j * (16 * HW);
    const int    wb = j * 16 + (hi ? 2 : 0);
    float ss = 0.f, tt = 0.f;
    #pragma unroll
    for (int kk = 0; kk < 4; ++kk) {
      float x0 = xj[kk * 4 * HW];
      float x1 = xj[(kk * 4 + 1) * HW];
      float w0 = wpad[wb + kk * 4];
      float w1 = wpad[wb + kk * 4 + 1];
      float w2 = wpad[wb + kk * 4 + 2];
      ss += x0 * w1 + x1 * w2;   // s[kb] + s[kb+1]
      tt += x0 * w0 + x1 * w1;   // t[kb] + t[kb+1]
    }
    ss += __shfl_xor(ss, 16);    // combine the two K-halves of the wave
    tt += __shfl_xor(tt, 16);
    totalT += tt;
    if (lane < 16) { Stile[wave][j][lane] = ss; Ttile[wave][j][lane] = tt; }
  }

  // ---- Pass B (L2 re-stream): WMMA in-tile scans, store g, accumulate loss ----
  // Carries: runS = sum_{j'<j} S,  suffix = totalT - sum_{j'<=j} T.
  float runS = 0.f, runTi = 0.f, lacc = 0.f;
  #pragma unroll 2
  for (int j = 0; j < 32; ++j) {
    float Sj = Stile[wave][j][n];
    float Tj = Ttile[wave][j][n];
    runTi += Tj;
    float seed = runS + (totalT - runTi);
    runS += Sj;

    v8f acc;
    #pragma unroll
    for (int i = 0; i < 8; ++i) acc[i] = seed;

    const float* xj = xA + (size_t)j * (16 * HW);
    const int    wb = j * 16 + (hi ? 2 : 0);
    #pragma unroll
    for (int kk = 0; kk < 4; ++kk) {
      float x0 = xj[kk * 4 * HW];
      float x1 = xj[(kk * 4 + 1) * HW];
      v2f bs, bt;
      bs.x = x0 * wpad[wb + kk * 4 + 1]; bs.y = x1 * wpad[wb + kk * 4 + 2];
      bt.x = x0 * wpad[wb + kk * 4];     bt.y = x1 * wpad[wb + kk * 4 + 1];
      acc = __builtin_amdgcn_wmma_f32_16x16x4_f32(false, aL[kk], false, bs,
                                                  (short)0, acc, false, false);
      acc = __builtin_amdgcn_wmma_f32_16x16x4_f32(false, aU[kk], false, bt,
                                                  (short)0, acc, false, false);
    }

    // D layout: VGPR i = rows (i, i+8); lanes 0-15 row i, lanes 16-31 row i+8.
    const float* xlj = xL + (size_t)j * (16 * HW);
    float*       glj = gL + (size_t)j * (16 * HW);
    #pragma unroll
    for (int i = 0; i < 8; ++i) {
      float xv = xlj[i * HW];
      float gv = acc[i];
      __builtin_nontemporal_store(gv, &glj[i * HW]);  // g is never re-read
      float d = gv - xv;
      lacc += d * d;
    }
  }

  // wave-local loss reduction, fixed order
  #pragma unroll
  for (int off = 16; off > 0; off >>= 1) lacc += __shfl_xor(lacc, off);
  if (lane == 0) partials[G] = lacc;
}

// Deterministic final reduction: fixed per-lane strided sums + fixed shuffle tree.
__global__ void loss_reduce_kernel(const float* __restrict__ partials,
                                   float* __restrict__ out_loss)
{
  int lane = threadIdx.x;  // launched as <<<1,32>>>
  float s = 0.f;
  for (int i = lane; i < PIXGROUPS; i += 32) s += partials[i];
  #pragma unroll
  for (int off = 16; off > 0; off >>= 1) s += __shfl_xor(s, off);
  if (lane == 0) *out_loss = sqrtf(s) * 0.0025f;
}

extern "C" void kernel_launch(void* const* d_in, const int* in_sizes, int n_in,
                              void* d_out, int out_size, void* d_ws, size_t ws_size,
                              hipStream_t stream) {
  const float* x = (const float*)d_in[0];   // (64, 512, 28, 28) f32
  const float* w = (const float*)d_in[1];   // (511,) f32
  float* g    = (float*)d_out;              // g flat, then 1 loss scalar
  float* loss = g + (size_t)BATCH * CH * HW;
  float* partials = (float*)d_ws;           // 3136 floats of scratch

  cooc_wmma_kernel<<<PIXGROUPS / WAVES_PER_BLOCK, 256, 0, stream>>>(x, w, g, partials);
  loss_reduce_kernel<<<1, 32, 0, stream>>>(partials, loss);
}